// L2SAGEConv_84859963834413
// MI455X (gfx1250) — compile-verified
//
#include <hip/hip_runtime.h>

// ---------------------------------------------------------------------------
// 2-layer GraphSAGE (mean aggr), fused for MI455X (gfx1250, wave32).
//
// Trick: layer-2 does  relu( mean_agg(h) @ w2_l.T + b2 + h @ w2_r.T ).
// Since deg-division is a per-node scalar, mean_agg(h) @ W.T == segsum(h@W.T)/deg,
// so we project h -> (p,q) [N,4] inside the layer-1 kernel and aggregate the
// 4-channel p over edges instead of the 400-channel h.  h never touches HBM.
// ---------------------------------------------------------------------------

#define N_NODES 50000
#define N_EDGES 800000
#define IN_CH   64
#define OUT_CH  4
#define HID     400   // 25 column tiles of 16

typedef __attribute__((ext_vector_type(2))) float v2f;
typedef __attribute__((ext_vector_type(8))) float v8f;

// ---------------------------------------------------------------------------
__global__ __launch_bounds__(256) void zero_ws_kernel(float* __restrict__ ws, int n) {
    int i = blockIdx.x * 256 + threadIdx.x;
    int stride = gridDim.x * 256;
    for (; i < n; i += stride) ws[i] = 0.0f;
}

// ---------------------------------------------------------------------------
// Layer-1 edge aggregation: one wave32 per edge; each lane moves 2 floats of
// the 64-float source row (one coalesced 256B row read), scatter-adds into
// aggsum[dst] (12.8MB accumulator -> lives in L2).  Lane 0 counts degree.
__global__ __launch_bounds__(256) void edge_agg1_kernel(
    const float* __restrict__ x, const int* __restrict__ ei,
    float* __restrict__ aggsum, float* __restrict__ deg) {
    const int gid  = blockIdx.x * 256 + threadIdx.x;
    const int e    = gid >> 5;
    const int lane = gid & 31;
    if (e >= N_EDGES) return;
    const int src = ei[e];
    const int dst = ei[N_EDGES + e];
    const float2 v = *(const float2*)(x + (size_t)src * IN_CH + lane * 2);
    float* a = aggsum + (size_t)dst * IN_CH + lane * 2;
    atomicAdd(a + 0, v.x);
    atomicAdd(a + 1, v.y);
    if (lane == 0) atomicAdd(deg + dst, 1.0f);
}

// ---------------------------------------------------------------------------
// Fused layer-1 + layer-2 projection.
//   h[16x400] = relu(agg@w1_l.T + b1 + x@w1_r.T)   via V_WMMA_F32_16X16X4_F32
//   p = h@w2_l.T, q = h@w2_r.T                      (only these hit HBM)
// One workgroup (8 waves) per 16-node row tile; waves split 25 column tiles.
// Both A fragments (agg/x) and B fragments (w1_l/w1_r) are fully register-
// resident before the 32-WMMA MAC chain, so VMEM latency is amortized once
// per column tile instead of per k-step.
__global__ __launch_bounds__(256) void sage_l1_fused_kernel(
    const float* __restrict__ x,    const float* __restrict__ aggsum,
    const float* __restrict__ deg,
    const float* __restrict__ w1l,  const float* __restrict__ b1,
    const float* __restrict__ w1r,
    const float* __restrict__ w2l,  const float* __restrict__ w2r,
    float* __restrict__ p, float* __restrict__ q) {

    __shared__ float aT[16 * 68];   // agg tile, row stride 68 (bank-conflict pad)
    __shared__ float xT[16 * 68];   // x tile
    __shared__ float hT[16 * 404];  // relu'd hidden tile, stride 404 (16B aligned rows)

    const int tid = threadIdx.x;
    const int nb  = blockIdx.x;

    // ---- Phase 1: stage A tiles (and apply mean = aggsum/deg) -------------
    {
        const int m    = tid >> 4;          // row 0..15
        const int c4   = (tid & 15) * 4;    // col group
        const int node = nb * 16 + m;
        const float ds = 1.0f / fmaxf(deg[node], 1.0f);
        const float4 av = *(const float4*)(aggsum + (size_t)node * IN_CH + c4);
        const float4 xv = *(const float4*)(x      + (size_t)node * IN_CH + c4);
        aT[m * 68 + c4 + 0] = av.x * ds;
        aT[m * 68 + c4 + 1] = av.y * ds;
        aT[m * 68 + c4 + 2] = av.z * ds;
        aT[m * 68 + c4 + 3] = av.w * ds;
        xT[m * 68 + c4 + 0] = xv.x;
        xT[m * 68 + c4 + 1] = xv.y;
        xT[m * 68 + c4 + 2] = xv.z;
        xT[m * 68 + c4 + 3] = xv.w;
    }
    __syncthreads();

    // ---- Phase 2: WMMA over 25 column tiles -------------------------------
    // ISA 16x16x4 f32 layouts (sec 7.12.2):
    //   A: lane%16 = M row; VGPR{0,1} hold K = 2*(lane/16)+{0,1}
    //   B: lane%16 = N col; VGPR{0,1} hold K = 2*(lane/16)+{0,1}
    //   C: N = lane%16; M = vgpr + 8*(lane/16)
    const int lane  = tid & 31;
    const int wv    = tid >> 5;
    const int col   = lane & 15;          // A-row / B-col / C-col index
    const int kh    = (lane >> 4) << 1;   // 0 or 2
    const int mbase = (lane >> 4) << 3;   // 0 or 8

    // A fragments: loaded once, reused for every column tile.
    v2f aA[16], aX[16];
    #pragma unroll
    for (int i = 0; i < 16; ++i) {
        aA[i] = *(const v2f*)(&aT[col * 68 + 4 * i + kh]);
        aX[i] = *(const v2f*)(&xT[col * 68 + 4 * i + kh]);
    }

    for (int ct = wv; ct < 25; ct += 8) {
        const int cb = ct << 4;

        // Batch-load all B fragments for this tile (issued back-to-back, one
        // amortized wait, instead of load->wait->wmma x32 serial chain).
        const float* wlBase = w1l + (cb + col) * IN_CH + kh;
        const float* wrBase = w1r + (cb + col) * IN_CH + kh;
        v2f bL[16], bR[16];
        #pragma unroll
        for (int i = 0; i < 16; ++i) {
            bL[i] = *(const v2f*)(wlBase + 4 * i);
            bR[i] = *(const v2f*)(wrBase + 4 * i);
        }

        v8f acc = {0.f, 0.f, 0.f, 0.f, 0.f, 0.f, 0.f, 0.f};
        #pragma unroll
        for (int i = 0; i < 16; ++i) {
            acc = __builtin_amdgcn_wmma_f32_16x16x4_f32(
                false, aA[i], false, bL[i], (short)0, acc, false, false);
            acc = __builtin_amdgcn_wmma_f32_16x16x4_f32(
                false, aX[i], false, bR[i], (short)0, acc, false, false);
        }

        const float bias = b1[cb + col];
        #pragma unroll
        for (int r = 0; r < 8; ++r) {
            hT[(mbase + r) * 404 + cb + col] = fmaxf(acc[r] + bias, 0.0f);
        }
    }
    __syncthreads();

    // ---- Phase 3: collapse h (16x400) -> p,q (16x4 each), float4-vectorized
    if (tid < 128) {
        const int mm    = tid >> 3;
        const int cc    = tid & 3;
        const int which = (tid >> 2) & 1;
        const float* w2 = which ? w2r : w2l;
        const float4* hrow = (const float4*)&hT[mm * 404];      // 404*4B % 16 == 0
        const float4* wrow = (const float4*)&w2[cc * HID];      // 400*4B % 16 == 0
        float s = 0.0f;
        #pragma unroll 4
        for (int k = 0; k < HID / 4; ++k) {
            const float4 hv = hrow[k];
            const float4 wv2 = wrow[k];
            s += hv.x * wv2.x + hv.y * wv2.y + hv.z * wv2.z + hv.w * wv2.w;
        }
        float* outp = which ? q : p;
        outp[(nb * 16 + mm) * 4 + cc] = s;
    }
}

// ---------------------------------------------------------------------------
// Layer-2 edge aggregation over the 4-channel projection p (12.8MB traffic
// instead of 1.28GB over h).
__global__ __launch_bounds__(256) void edge_agg2_kernel(
    const float* __restrict__ p, const int* __restrict__ ei,
    float* __restrict__ aggp) {
    const int e = blockIdx.x * 256 + threadIdx.x;
    if (e >= N_EDGES) return;
    const int src = ei[e];
    const int dst = ei[N_EDGES + e];
    const float4 v = *(const float4*)(p + (size_t)src * 4);
    float* a = aggp + (size_t)dst * 4;
    atomicAdd(a + 0, v.x);
    atomicAdd(a + 1, v.y);
    atomicAdd(a + 2, v.z);
    atomicAdd(a + 3, v.w);
}

// ---------------------------------------------------------------------------
__global__ __launch_bounds__(256) void sage_l2_final_kernel(
    const float* __restrict__ aggp, const float* __restrict__ deg,
    const float* __restrict__ b2,   const float* __restrict__ q,
    float* __restrict__ out) {
    const int i = blockIdx.x * 256 + threadIdx.x;
    if (i >= N_NODES) return;
    const float ds = 1.0f / fmaxf(deg[i], 1.0f);
    const float4 a  = *(const float4*)(aggp + (size_t)i * 4);
    const float4 qq = *(const float4*)(q    + (size_t)i * 4);
    float4 o;
    o.x = fmaxf(a.x * ds + b2[0] + qq.x, 0.0f);
    o.y = fmaxf(a.y * ds + b2[1] + qq.y, 0.0f);
    o.z = fmaxf(a.z * ds + b2[2] + qq.z, 0.0f);
    o.w = fmaxf(a.w * ds + b2[3] + qq.w, 0.0f);
    *(float4*)(out + (size_t)i * 4) = o;
}

// ---------------------------------------------------------------------------
extern "C" void kernel_launch(void* const* d_in, const int* in_sizes, int n_in,
                              void* d_out, int out_size, void* d_ws, size_t ws_size,
                              hipStream_t stream) {
    const float* x   = (const float*)d_in[0];
    const int*   ei  = (const int*)d_in[1];   // [2, E] int32 (JAX x64 disabled)
    const float* w1l = (const float*)d_in[2];
    const float* b1  = (const float*)d_in[3];
    const float* w1r = (const float*)d_in[4];
    const float* w2l = (const float*)d_in[5];
    const float* b2  = (const float*)d_in[6];
    const float* w2r = (const float*)d_in[7];
    float* out = (float*)d_out;

    // Workspace layout (floats): aggsum[N*64] | deg[N] | p[N*4] | q[N*4] | aggp[N*4]
    float* ws     = (float*)d_ws;
    float* aggsum = ws;
    float* deg    = aggsum + (size_t)N_NODES * IN_CH;
    float* p      = deg    + N_NODES;
    float* q      = p      + (size_t)N_NODES * 4;
    float* aggp   = q      + (size_t)N_NODES * 4;
    const int ws_floats = N_NODES * IN_CH + N_NODES + 3 * N_NODES * 4;  // ~15.4 MB

    zero_ws_kernel<<<2048, 256, 0, stream>>>(ws, ws_floats);

    edge_agg1_kernel<<<(N_EDGES * 32) / 256, 256, 0, stream>>>(x, ei, aggsum, deg);

    sage_l1_fused_kernel<<<N_NODES / 16, 256, 0, stream>>>(
        x, aggsum, deg, w1l, b1, w1r, w2l, w2r, p, q);

    edge_agg2_kernel<<<(N_EDGES + 255) / 256, 256, 0, stream>>>(p, ei, aggp);

    sage_l2_final_kernel<<<(N_NODES + 255) / 256, 256, 0, stream>>>(
        aggp, deg, b2, q, out);
}